// GlimpseModel_90787018703705
// MI455X (gfx1250) — compile-verified
//
#include <hip/hip_runtime.h>

// Problem constants (fixed by the reference)
#define IMG_H 1024
#define IMG_W 1024
#define NKER  144
#define NTILE 9        // 144 / 16 kernel tiles
#define MAXWIN 256     // union window cap (analysis: <= ~165 px needed)
#define BLOCK 128
#define NWAVES 4

typedef __attribute__((ext_vector_type(2))) float v2f;
typedef __attribute__((ext_vector_type(8))) float v8f;

__global__ __launch_bounds__(BLOCK)
void glimpse_wmma_kernel(const float* __restrict__ imgs,
                         const float* __restrict__ s_c,
                         const float* __restrict__ s_z,
                         const float* __restrict__ mu,
                         const float* __restrict__ sigma,
                         float* __restrict__ out)
{
    // Column (fx) and row (fy) Gaussian weights over the union window, [t][k]
    __shared__ float fxLDS[MAXWIN * 16];
    __shared__ float fyLDS[MAXWIN * 16];
    __shared__ float m0s[16], m1s[16], cs[16], Zs[16];
    __shared__ int   jloK[16], jhiK[16], iloK[16], ihiK[16];
    __shared__ int   uni[4];           // jlo, jcount, ilo, icount
    __shared__ float part[BLOCK];      // per-lane partial sums

    const int tid  = threadIdx.x;
    const int lane = tid & 31;
    const int wave = tid >> 5;
    const int h    = lane >> 4;        // lane half (0/1)
    const int kl   = lane & 15;        // kernel column within tile
    const int kt   = blockIdx.x;       // kernel tile 0..8
    const int b    = blockIdx.y;       // batch

    const float sz = s_z[b];

    // --- Per-kernel parameters + exact-zero windows (exp underflow bound) ---
    if (tid < 16) {
        const int kg = kt * 16 + tid;
        const float m0 = (s_c[b * 2 + 0] + mu[kg * 2 + 0]) * sz; // column center
        const float m1 = (s_c[b * 2 + 1] + mu[kg * 2 + 1]) * sz; // row center
        const float sg = sigma[kg] * sz;
        const float c  = 0.5f / (sg * sg);
        // exponent at radius = (14.9)^2/2 = 111 > 103.9 -> expf == +0.0f exactly
        const float rad = sg * 14.9f;
        m0s[tid] = m0; m1s[tid] = m1; cs[tid] = c;
        jloK[tid] = max(0,         (int)floorf((m0 - rad + 1.0f) * 511.5f));
        jhiK[tid] = min(IMG_W - 1, (int)ceilf ((m0 + rad + 1.0f) * 511.5f));
        iloK[tid] = max(0,         (int)floorf((m1 - rad + 1.0f) * 511.5f));
        ihiK[tid] = min(IMG_H - 1, (int)ceilf ((m1 + rad + 1.0f) * 511.5f));
    }
    __syncthreads();

    if (tid == 0) {
        int jl = IMG_W - 1, jh = 0, il = IMG_H - 1, ih = 0;
        for (int k = 0; k < 16; ++k) {
            jl = min(jl, jloK[k]); jh = max(jh, jhiK[k]);
            il = min(il, iloK[k]); ih = max(ih, ihiK[k]);
        }
        // Align start down (mult of 4 -> 8B-aligned float2 A-loads), pad counts.
        jl &= ~3;
        int jc = ((jh - jl + 1) + 3)  & ~3;   if (jc > MAXWIN) jc = MAXWIN;
        int ic = ((ih - il + 1) + 15) & ~15;  if (ic > MAXWIN) ic = MAXWIN;
        // Shift window so padded extent stays in-bounds: extra low-side
        // entries are beyond every kernel's radius -> weight == +0.0f exactly.
        if (jl + jc > IMG_W) jl = IMG_W - jc;   // stays mult of 4 (W mult of 4)
        if (il + ic > IMG_H) il = IMG_H - ic;
        uni[0] = jl; uni[1] = jc; uni[2] = il; uni[3] = ic;
    }
    __syncthreads();

    const int jlo = uni[0], jcount = uni[1], ilo = uni[2], icount = uni[3];
    const float STEP = 2.0f / 1023.0f;   // linspace(-1,1,1024) step

    // --- Fill separable Gaussian factors into LDS.
    //     Entries beyond the true union window underflow to exact +0.0f. ---
    for (int idx = tid; idx < MAXWIN * 16; idx += BLOCK) {
        const int t = idx >> 4, k = idx & 15;
        const float dj = fmaf((float)(jlo + t), STEP, -1.0f) - m0s[k];
        fxLDS[t * 16 + k] = expf(-(dj * dj) * cs[k]);
        const float di = fmaf((float)(ilo + t), STEP, -1.0f) - m1s[k];
        fyLDS[t * 16 + k] = expf(-(di * di) * cs[k]);
    }
    __syncthreads();

    // --- Normalizer Z[k] = (sum fy)*(sum fx) + 1e-7 (windowed == full, exact zeros) ---
    if (tid < 16) {
        float sx = 0.0f, sy = 0.0f;
        for (int t = 0; t < MAXWIN; ++t) {
            sx += fxLDS[t * 16 + tid];
            sy += fyLDS[t * 16 + tid];
        }
        Zs[tid] = sy * sx + 1e-7f;
    }
    __syncthreads();

    // --- Main contraction: S[i,k] = sum_j imgs[i,j]*fx[k,j] via V_WMMA_F32_16X16X4_F32,
    //     then out_k += sum_i fy[k,i]*S[i,k]. 16-row chunks round-robin over waves. ---
    const float* __restrict__ imgb = imgs + (size_t)b * IMG_H * IMG_W;
    const int nchunks = icount >> 4;
    float pacc = 0.0f;

    for (int cc = wave; cc < nchunks; cc += NWAVES) {
        v8f S = {};
        // A layout (16x4 f32, MxK): lane holds row M = lane&15, VGPRs K = {2h, 2h+1}
        const int irow = ilo + cc * 16 + kl;            // always in-bounds (shifted window)
        const float* __restrict__ p = imgb + (size_t)irow * IMG_W + jlo + 2 * h;
        __builtin_prefetch(p, 0, 0);                    // global_prefetch_b8: warm row window
        // B layout (4x16 f32, KxN): lane holds col N = lane&15, VGPRs K = {2h, 2h+1}
        int ldsoff = (2 * h) * 16 + kl;

        for (int jloc = 0; jloc < jcount; jloc += 4) {
            const v2f a = *(const v2f*)p;               // 8B-aligned global_load_b64
            p += 4;
            v2f bv;
            bv.x = fxLDS[ldsoff];
            bv.y = fxLDS[ldsoff + 16];
            ldsoff += 64;
            S = __builtin_amdgcn_wmma_f32_16x16x4_f32(
                    /*neg_a=*/false, a, /*neg_b=*/false, bv,
                    /*c_mod=*/(short)0, S, /*reuse_a=*/false, /*reuse_b=*/false);
        }
        // C layout: lane holds N = lane&15, VGPR r -> row M = r + 8h
#pragma unroll
        for (int r = 0; r < 8; ++r) {
            const int iloc = cc * 16 + r + 8 * h;
            pacc += fyLDS[iloc * 16 + kl] * S[r];
        }
    }

    part[tid] = pacc;
    __syncthreads();

    // --- Final reduce: both lane halves x 4 waves, then normalize ---
    if (tid < 16) {
        float tot = 0.0f;
        for (int w = 0; w < NWAVES; ++w)
            tot += part[w * 32 + tid] + part[w * 32 + 16 + tid];
        out[(size_t)b * NKER + kt * 16 + tid] = tot / Zs[tid];
    }
}

extern "C" void kernel_launch(void* const* d_in, const int* in_sizes, int n_in,
                              void* d_out, int out_size, void* d_ws, size_t ws_size,
                              hipStream_t stream) {
    const float* imgs  = (const float*)d_in[0];
    const float* s_c   = (const float*)d_in[1];
    const float* s_z   = (const float*)d_in[2];
    const float* mu    = (const float*)d_in[3];
    const float* sigma = (const float*)d_in[4];
    float* out = (float*)d_out;

    const int B = in_sizes[2];          // s_z has B elements
    dim3 grid(NTILE, B);
    glimpse_wmma_kernel<<<grid, BLOCK, 0, stream>>>(imgs, s_c, s_z, mu, sigma, out);
}